// Net_70617852280963
// MI455X (gfx1250) — compile-verified
//
#include <hip/hip_runtime.h>
#include <hip/hip_bf16.h>

// RGCN 2-layer forward for MI455X (gfx1250, wave32).
// Sizes fixed by the reference:
#define NN 50000
#define RR 46
#define CC 4
#define BB 30
#define HH 16
#define NH (NN * HH)          // 800000 columns of the W1 GEMM

typedef __attribute__((ext_vector_type(2))) float v2f;
typedef __attribute__((ext_vector_type(8))) float v8f;

// ---------------------------------------------------------------------------
// K0: init  h[n,i] = root1[n,i] + bias1[i];  cnt = 0;  out = 0
// ---------------------------------------------------------------------------
__global__ void k_init(const float* __restrict__ root1,
                       const float* __restrict__ bias1,
                       unsigned* __restrict__ cnt,
                       float* __restrict__ h,
                       float* __restrict__ out)
{
    int t = blockIdx.x * blockDim.x + threadIdx.x;
    if (t < NN * RR) cnt[t] = 0u;
    if (t < NN * HH) h[t] = root1[t] + bias1[t & (HH - 1)];
    if (t < NN * CC) out[t] = 0.0f;
}

// ---------------------------------------------------------------------------
// K1: per-(dst, rel) edge counts (streaming; prefetch ahead on both streams)
// ---------------------------------------------------------------------------
__global__ void k_count(const int* __restrict__ dst,
                        const int* __restrict__ rel,
                        unsigned* __restrict__ cnt, int E)
{
    int e = blockIdx.x * blockDim.x + threadIdx.x;
    if (e >= E) return;
    // Speculative prefetch (ISA 10.5: translation faults silently dropped)
    __builtin_prefetch(dst + e + 8192, 0, 0);
    __builtin_prefetch(rel + e + 8192, 0, 0);
    atomicAdd(&cnt[dst[e] * RR + rel[e]], 1u);
}

// ---------------------------------------------------------------------------
// K2: W1[r, col] = sum_b comp1[r,b] * basis1[b, col]     (col = n*H + i)
// One wave per 16-column tile of the [48pad, 800000] output; each wave
// accumulates all 3 relation row-tiles so the basis1 fragment is reused 3x.
// K = 30 padded to 32 -> 8 x V_WMMA_F32_16X16X4_F32 per row tile.
//
// comp1 is staged zero-padded [48][32] in LDS so every A-fragment read is a
// branch-free ds_load_b64 (no EXEC divergence anywhere near the WMMAs).
// The K tail (rows 30,31) is handled with unconditional loads of rows 28,29
// (in-bounds for every lane) + v_cndmask zero-select for the upper half.
//
// ISA layouts (7.12.2):
//   A 16x4 f32 : lanes 0-15 M=lane, {K=k0,k0+1}; lanes 16-31 M=lane-16, {K=k0+2,k0+3}
//   B 4x16 f32 : lanes 0-15 {K=k0,k0+1}, lanes 16-31 {K=k0+2,k0+3}; N = lane&15
//   C/D 16x16  : VGPR v, lanes 0-15 M=v; lanes 16-31 M=v+8; N = lane&15
// ---------------------------------------------------------------------------
__global__ __launch_bounds__(128) void k_w1_gemm(
        const float* __restrict__ comp1,   // [R, B] row-major
        const float* __restrict__ basis1,  // [B, N*H] row-major
        float* __restrict__ W1)            // [R, N*H] row-major
{
    __shared__ float lA[48 * 32];          // comp1 zero-padded [m][k], 6 KB

    for (int idx = threadIdx.x; idx < 48 * 32; idx += 128) {
        int m = idx >> 5;
        int k = idx & 31;
        lA[idx] = (m < RR && k < BB) ? comp1[m * BB + k] : 0.0f;
    }
    __syncthreads();

    const int wave = (blockIdx.x << 2) + (threadIdx.x >> 5);  // 0..49999
    const int lane = threadIdx.x & 31;
    const int half = lane >> 4;     // 0: lanes 0-15, 1: lanes 16-31
    const int l    = lane & 15;
    const int col  = wave * 16 + l; // this lane's output column

    v8f acc0 = {}, acc1 = {}, acc2 = {};

    #pragma unroll
    for (int ks = 0; ks < 8; ++ks) {
        const int k = ks * 4 + half * 2;          // this lane's K pair (even)

        // B fragment: basis1[k, col], basis1[k+1, col]
        v2f bf;
        if (ks < 7) {                              // K <= 27: always in-bounds
            bf.x = basis1[(size_t)k * NH + col];
            bf.y = basis1[(size_t)(k + 1) * NH + col];
        } else {                                   // tail: rows 28..31
            float b28 = basis1[(size_t)28 * NH + col];  // safe for all lanes
            float b29 = basis1[(size_t)29 * NH + col];
            bf.x = half ? 0.0f : b28;              // rows 30,31 are zero pad
            bf.y = half ? 0.0f : b29;
        }

        // A fragments from padded LDS: branch-free 8B loads
        v2f a0 = *(const v2f*)&lA[(0 * 16 + l) * 32 + k];
        v2f a1 = *(const v2f*)&lA[(1 * 16 + l) * 32 + k];
        v2f a2 = *(const v2f*)&lA[(2 * 16 + l) * 32 + k];

        acc0 = __builtin_amdgcn_wmma_f32_16x16x4_f32(false, a0, false, bf,
                                                     (short)0, acc0, false, false);
        acc1 = __builtin_amdgcn_wmma_f32_16x16x4_f32(false, a1, false, bf,
                                                     (short)0, acc1, false, false);
        acc2 = __builtin_amdgcn_wmma_f32_16x16x4_f32(false, a2, false, bf,
                                                     (short)0, acc2, false, false);
    }

    // Store D tiles: element v of acc -> row m = rt*16 + v + half*8
    #pragma unroll
    for (int v = 0; v < 8; ++v) {
        int m0 = 0 * 16 + v + half * 8;
        W1[(size_t)m0 * NH + col] = acc0[v];
        int m1 = 1 * 16 + v + half * 8;
        W1[(size_t)m1 * NH + col] = acc1[v];
        int m2 = 2 * 16 + v + half * 8;          // 32..47: rows 46,47 dropped
        if (m2 < RR) W1[(size_t)m2 * NH + col] = acc2[v];
    }
}

// ---------------------------------------------------------------------------
// K3: W2t[r, c, i] = sum_b comp2[r,b] * basis2[b, i, c]   (transposed for K5)
// ---------------------------------------------------------------------------
__global__ void k_w2(const float* __restrict__ comp2,   // [R, B]
                     const float* __restrict__ basis2,  // [B, H, C]
                     float* __restrict__ W2t)           // [R, C, H]
{
    int t = blockIdx.x * blockDim.x + threadIdx.x;      // R*H*C = 2944
    if (t >= RR * HH * CC) return;
    int r = t / (HH * CC);
    int rem = t - r * (HH * CC);
    int c = rem / HH;
    int i = rem - c * HH;
    float acc = 0.0f;
    #pragma unroll
    for (int b = 0; b < BB; ++b)
        acc += comp2[r * BB + b] * basis2[b * (HH * CC) + i * CC + c];
    W2t[r * (CC * HH) + c * HH + i] = acc;
}

// ---------------------------------------------------------------------------
// K4: layer-1 edge scatter. thread = (edge, i); 16 consecutive threads read a
// contiguous 64B row of W1 -> coalesced gathers; atomics land in L2 (h: 3.2MB)
// ---------------------------------------------------------------------------
__global__ void k_edge1(const int* __restrict__ src,
                        const int* __restrict__ dst,
                        const int* __restrict__ rel,
                        const unsigned* __restrict__ cnt,
                        const float* __restrict__ W1,
                        float* __restrict__ h, int E)
{
    int t = blockIdx.x * blockDim.x + threadIdx.x;
    if (t >= E * HH) return;
    int e = t >> 4;
    int i = t & 15;
    int s = src[e], d = dst[e], r = rel[e];
    float inv = 1.0f / (float)cnt[d * RR + r];
    float w = W1[(size_t)r * NH + s * HH + i];
    atomicAdd(&h[d * HH + i], w * inv);
}

// ---------------------------------------------------------------------------
// K5: layer-2 edge scatter. thread = (edge, c); ReLU applied on the gather.
// ---------------------------------------------------------------------------
__global__ void k_edge2(const int* __restrict__ src,
                        const int* __restrict__ dst,
                        const int* __restrict__ rel,
                        const unsigned* __restrict__ cnt,
                        const float* __restrict__ h,
                        const float* __restrict__ W2t,   // [R, C, H]
                        float* __restrict__ out, int E)
{
    int t = blockIdx.x * blockDim.x + threadIdx.x;
    if (t >= E * CC) return;
    int e = t >> 2;
    int c = t & 3;
    int s = src[e], d = dst[e], r = rel[e];
    float inv = 1.0f / (float)cnt[d * RR + r];
    const float* hv = h + s * HH;
    const float* wv = W2t + r * (CC * HH) + c * HH;
    float acc = 0.0f;
    #pragma unroll
    for (int i = 0; i < HH; ++i)
        acc += fmaxf(hv[i], 0.0f) * wv[i];
    atomicAdd(&out[d * CC + c], acc * inv);
}

// ---------------------------------------------------------------------------
// K6: out[n,:] += relu(h[n,:]) @ root2 + bias2 ; then log_softmax over C=4
// ---------------------------------------------------------------------------
__global__ void k_final(const float* __restrict__ h,
                        const float* __restrict__ root2,  // [H, C]
                        const float* __restrict__ bias2,  // [C]
                        float* __restrict__ out)
{
    int n = blockIdx.x * blockDim.x + threadIdx.x;
    if (n >= NN) return;
    float acc[CC];
    #pragma unroll
    for (int c = 0; c < CC; ++c) acc[c] = out[n * CC + c] + bias2[c];
    #pragma unroll
    for (int i = 0; i < HH; ++i) {
        float hv = fmaxf(h[n * HH + i], 0.0f);
        #pragma unroll
        for (int c = 0; c < CC; ++c) acc[c] += hv * root2[i * CC + c];
    }
    float m = acc[0];
    #pragma unroll
    for (int c = 1; c < CC; ++c) m = fmaxf(m, acc[c]);
    float s = 0.0f;
    #pragma unroll
    for (int c = 0; c < CC; ++c) s += expf(acc[c] - m);
    float ls = m + logf(s);
    #pragma unroll
    for (int c = 0; c < CC; ++c) out[n * CC + c] = acc[c] - ls;
}

// ---------------------------------------------------------------------------
extern "C" void kernel_launch(void* const* d_in, const int* in_sizes, int n_in,
                              void* d_out, int out_size, void* d_ws, size_t ws_size,
                              hipStream_t stream)
{
    // Inputs (setup_inputs order)
    const int*   edge_index = (const int*)d_in[0];   // [2, E]
    const int*   edge_type  = (const int*)d_in[1];   // [E]
    const float* basis1     = (const float*)d_in[2]; // [B, N, H]
    const float* comp1      = (const float*)d_in[3]; // [R, B]
    const float* root1      = (const float*)d_in[4]; // [N, H]
    const float* bias1      = (const float*)d_in[5]; // [H]
    const float* basis2     = (const float*)d_in[6]; // [B, H, C]
    const float* comp2      = (const float*)d_in[7]; // [R, B]
    const float* root2      = (const float*)d_in[8]; // [H, C]
    const float* bias2      = (const float*)d_in[9]; // [C]
    float* out = (float*)d_out;                      // [N, C]

    const int E = in_sizes[1];
    const int* src = edge_index;
    const int* dst = edge_index + E;

    // Workspace carve-up (256B aligned)
    size_t off = 0;
    auto carve = [&](size_t bytes) {
        size_t o = off;
        off = (off + bytes + 255) & ~(size_t)255;
        return o;
    };
    unsigned* cnt = (unsigned*)((char*)d_ws + carve((size_t)NN * RR * 4)); // 9.2 MB
    float*    h   = (float*)   ((char*)d_ws + carve((size_t)NN * HH * 4)); // 3.2 MB
    float*    W2t = (float*)   ((char*)d_ws + carve((size_t)RR * CC * HH * 4));
    float*    W1  = (float*)   ((char*)d_ws + carve((size_t)RR * NH * 4)); // 147 MB
    (void)ws_size;

    const int TB = 256;

    // K0: init h / cnt / out
    k_init<<<(NN * RR + TB - 1) / TB, TB, 0, stream>>>(root1, bias1, cnt, h, out);

    // K1: per-(dst,rel) counts
    k_count<<<(E + TB - 1) / TB, TB, 0, stream>>>(dst, edge_type, cnt, E);

    // K2: W1 = comp1 @ basis1 via WMMA (NH/16 = 50000 column tiles, 4 waves/block)
    k_w1_gemm<<<(NH / 16) / 4, 128, 0, stream>>>(comp1, basis1, W1);

    // K3: W2 (transposed)
    k_w2<<<(RR * HH * CC + TB - 1) / TB, TB, 0, stream>>>(comp2, basis2, W2t);

    // K4: layer-1 scatter
    k_edge1<<<((size_t)E * HH + TB - 1) / TB, TB, 0, stream>>>(src, dst, edge_type,
                                                               cnt, W1, h, E);

    // K5: layer-2 scatter
    k_edge2<<<((size_t)E * CC + TB - 1) / TB, TB, 0, stream>>>(src, dst, edge_type,
                                                               cnt, h, W2t, out, E);

    // K6: dense term + log_softmax
    k_final<<<(NN + TB - 1) / TB, TB, 0, stream>>>(h, root2, bias2, out);
}